// YuKeMLPFLOW_onlyX_seperateZ_init_37460704756279
// MI455X (gfx1250) — compile-verified
//
#include <hip/hip_runtime.h>
#include <math.h>

typedef float v2f __attribute__((ext_vector_type(2)));
typedef float v4f __attribute__((ext_vector_type(4)));
typedef float v8f __attribute__((ext_vector_type(8)));

#define B_N   8192
#define L_N   12
#define H_N   64
#define SLOPE 0.2f
#define EPS_F 1e-6f
#define ROWS  16
#define LSTR  68   // LDS row stride in floats: (68*m + k) % 64 = (4m + k) % 64 -> conflict-free

__device__ __forceinline__ float leaky(float z)  { return z >= 0.f ? z : SLOPE * z; }
__device__ __forceinline__ float dleaky(float z) { return z >= 0.f ? 1.f : SLOPE; }

// ---------------- Kernel 0: stream-fill logdet with L*log(eps) ----------------
__global__ __launch_bounds__(256) void fill_logdet_kernel(float* __restrict__ logdet,
                                                          long long n4, float val) {
    v4f v = {val, val, val, val};
    v4f* p = (v4f*)logdet;
    long long i = (long long)blockIdx.x * blockDim.x + threadIdx.x;
    long long stride = (long long)gridDim.x * blockDim.x;
    for (; i < n4; i += stride)
        __builtin_nontemporal_store(v, p + i);   // 256MB stream; don't thrash L2
}

// ---------------- Kernel 1: fused 3-layer MLP + JVP, one wave per (16-row tile, flow) ----
__global__ __launch_bounds__(256) void flow_kernel(
    const float* __restrict__ x,  const float* __restrict__ W1, const float* __restrict__ b1,
    const float* __restrict__ W2, const float* __restrict__ b2,
    const float* __restrict__ W3, const float* __restrict__ b3,
    const float* __restrict__ W4, const float* __restrict__ b4,
    float* __restrict__ residuals, float* __restrict__ dout)
{
    __shared__ float smem[8 * 2 * ROWS * LSTR];   // per-wave a/t staging, 69632 B / block

    const int wave = threadIdx.x >> 5;
    const int lane = threadIdx.x & 31;
    const int lm   = lane & 15;     // N / M index within half-wave
    const int lh   = lane >> 4;     // half-wave select
    const int unit = blockIdx.x * 8 + wave;       // 6144 units total
    const int flow = unit % L_N;
    const int r0   = (unit / L_N) * ROWS;

    float* aBuf = smem + wave * (2 * ROWS * LSTR);
    float* tBuf = aBuf + ROWS * LSTR;

    const float* Wl2 = W2 + flow * H_N * H_N;
    const float* Wl3 = W3 + flow * H_N * H_N;

    // ---- Layer 1: z1 = x*W1 + b1 (outer product), built directly in C/D fragment layout
    float xv[8];
#pragma unroll
    for (int g = 0; g < 8; ++g)
        xv[g] = x[(r0 + g + 8 * lh) * L_N + flow];

    v8f aF[4], tF[4];   // activation / tangent fragments: 4 N-tiles of 16x16
#pragma unroll
    for (int n = 0; n < 4; ++n) {
        const int col = n * 16 + lm;
        const float w  = W1[flow * H_N + col];
        const float bb = b1[flow * H_N + col];
#pragma unroll
        for (int g = 0; g < 8; ++g) {
            float z = xv[g] * w + bb;
            aF[n][g] = leaky(z);
            tF[n][g] = w * dleaky(z);
        }
    }

    // ---- Layers 2 and 3: [16,64] @ [64,64] via V_WMMA_F32_16X16X4_F32, shared B frags
#pragma unroll
    for (int layer = 0; layer < 2; ++layer) {
        const float* Wl = (layer == 0) ? Wl2 : Wl3;
        const float* bl = (layer == 0) ? b2 : b3;

        // stage current a/t fragments to LDS (C-layout: row = g + 8*lh, col = n*16+lm)
#pragma unroll
        for (int n = 0; n < 4; ++n) {
            const int col = n * 16 + lm;
#pragma unroll
            for (int g = 0; g < 8; ++g) {
                const int row = g + 8 * lh;
                aBuf[row * LSTR + col] = aF[n][g];
                tBuf[row * LSTR + col] = tF[n][g];
            }
        }
        __syncthreads();

        // accumulators: z = bias, tz = 0
#pragma unroll
        for (int n = 0; n < 4; ++n) {
            const float bb = bl[flow * H_N + n * 16 + lm];
#pragma unroll
            for (int g = 0; g < 8; ++g) { aF[n][g] = bb; tF[n][g] = 0.f; }
        }

        for (int k0 = 0; k0 < H_N; k0 += 4) {
            // A fragments: lane lm = row M, half-wave picks K pair {k0+2lh, k0+2lh+1}
            const int abase = lm * LSTR + k0 + 2 * lh;
            v2f Aa, At;
            Aa.x = aBuf[abase]; Aa.y = aBuf[abase + 1];
            At.x = tBuf[abase]; At.y = tBuf[abase + 1];

            const float* wp = Wl + (k0 + 2 * lh) * H_N + lm;
#pragma unroll
            for (int n = 0; n < 4; ++n) {
                v2f Bf;                       // B[k][col]: VGPR0 = K {0,2}, VGPR1 = K {1,3}
                Bf.x = wp[n * 16];
                Bf.y = wp[n * 16 + H_N];
                aF[n] = __builtin_amdgcn_wmma_f32_16x16x4_f32(
                            false, Aa, false, Bf, (short)0, aF[n], false, false);
                tF[n] = __builtin_amdgcn_wmma_f32_16x16x4_f32(
                            false, At, false, Bf, (short)0, tF[n], false, false);
            }
        }

        // activation + tangent gating, elementwise on fragments
#pragma unroll
        for (int n = 0; n < 4; ++n)
#pragma unroll
            for (int g = 0; g < 8; ++g) {
                float z = aF[n][g];
                aF[n][g] = leaky(z);
                tF[n][g] = tF[n][g] * dleaky(z);
            }
        __syncthreads();
    }

    // ---- Output layer: stage a3/t3, then lanes 0-15 do residual rows, lanes 16-31 do dout rows
#pragma unroll
    for (int n = 0; n < 4; ++n) {
        const int col = n * 16 + lm;
#pragma unroll
        for (int g = 0; g < 8; ++g) {
            const int row = g + 8 * lh;
            aBuf[row * LSTR + col] = aF[n][g];
            tBuf[row * LSTR + col] = tF[n][g];
        }
    }
    __syncthreads();

    const float* src = (lane < 16) ? aBuf : tBuf;
    const float* w4  = W4 + flow * H_N;
    float acc = 0.f;
#pragma unroll 8
    for (int e = 0; e < H_N; ++e)
        acc += src[lm * LSTR + e] * w4[e];

    if (lane < 16)
        residuals[(r0 + lm) * L_N + flow] = acc + b4[flow];
    else
        dout[(r0 + lm) * L_N + flow] = acc;   // raw JVP; reduced deterministically in K2
}

// ---------------- Kernel 2: diagonal patch: logdet[i,i] = sum_l log(|dout|+eps) ----------
__global__ __launch_bounds__(256) void diag_kernel(const float* __restrict__ dout,
                                                   float* __restrict__ logdet) {
    const int i = blockIdx.x * blockDim.x + threadIdx.x;   // 8192 threads
    float s = 0.f;
#pragma unroll
    for (int l = 0; l < L_N; ++l)
        s += logf(fabsf(dout[i * L_N + l]) + EPS_F);
    logdet[(long long)i * B_N + i] = s;
}

extern "C" void kernel_launch(void* const* d_in, const int* in_sizes, int n_in,
                              void* d_out, int out_size, void* d_ws, size_t ws_size,
                              hipStream_t stream) {
    const float* x  = (const float*)d_in[0];
    const float* W1 = (const float*)d_in[1];
    const float* b1 = (const float*)d_in[2];
    const float* W2 = (const float*)d_in[3];
    const float* b2 = (const float*)d_in[4];
    const float* W3 = (const float*)d_in[5];
    const float* b3 = (const float*)d_in[6];
    const float* W4 = (const float*)d_in[7];
    const float* b4 = (const float*)d_in[8];

    float* residuals = (float*)d_out;
    float* logdet    = residuals + (size_t)B_N * L_N;
    float* dout      = (float*)d_ws;          // B_N * L_N floats of scratch

    const float fillv = (float)(L_N * -13.815510557964274);   // L * log(1e-6)

    const long long n4 = (long long)B_N * B_N / 4;
    fill_logdet_kernel<<<4096, 256, 0, stream>>>(logdet, n4, fillv);

    const int units  = (B_N / ROWS) * L_N;    // 512 tiles * 12 flows = 6144 waves
    flow_kernel<<<units / 8, 256, 0, stream>>>(x, W1, b1, W2, b2, W3, b3, W4, b4,
                                               residuals, dout);

    diag_kernel<<<B_N / 256, 256, 0, stream>>>(dout, logdet);
}